// RandomGaussianTokenExtractor_19464791785491
// MI455X (gfx1250) — compile-verified
//
#include <hip/hip_runtime.h>

// Problem constants (match the reference).
#define B_   32
#define N_   256
#define C_   3
#define H_   512
#define W_   512
#define K_   32
#define ELEMS (C_ * K_ * K_)      // 3072 floats = 12 KB per token
#define THREADS 256
#define TOK 4                     // tokens per block (double-buffered pipeline)
#define NBLK (B_ * N_ / TOK)      // 2048 workgroups

typedef __attribute__((ext_vector_type(4))) unsigned int tdm_u32x4;
typedef __attribute__((ext_vector_type(8))) int          tdm_i32x8;
typedef __attribute__((ext_vector_type(4))) int          tdm_i32x4;

// Issue one TDM descriptor: DMA the (C,K,K) strided tile starting at
// img[b, 0, mu0, mu1] into LDS at byte offset lds_base.
__device__ __forceinline__ void tdm_issue(const float* img, int b, int mu0, int mu1,
                                          unsigned lds_base)
{
    unsigned long long ga = (unsigned long long)(uintptr_t)img +
        4ull * ((unsigned long long)b * (C_ * H_ * W_) +
                (unsigned long long)mu0 * W_ + (unsigned)mu1);

    // D# group 0: count=1 | lds_addr | global_addr | type=2 ("image")
    tdm_u32x4 g0;
    g0[0] = 1u;
    g0[1] = lds_base;
    g0[2] = (unsigned)(ga & 0xFFFFFFFFull);
    g0[3] = ((unsigned)(ga >> 32) & 0x01FFFFFFu) | 0x80000000u;

    // D# group 1: data_size=4B; tensor 512x512; tile 32x32x3;
    // dim0_stride=W (row), dim1_stride=H*W (channel; z uses dim1_stride in 3D).
    tdm_i32x8 g1;
    g1[0] = 0x00020000;                                     // data_size=2 (4B)
    g1[1] = (int)((unsigned)W_ << 16);                      // tensor_dim0 lo16
    g1[2] = (int)((unsigned)H_ << 16);                      // dim0 hi | tensor_dim1 lo
    g1[3] = (int)((unsigned)K_ << 16);                      // dim1 hi | tile_dim0=32
    g1[4] = (int)((unsigned)K_ | ((unsigned)C_ << 16));     // tile_dim1=32 | tile_dim2=3
    g1[5] = W_;                                             // tensor_dim0_stride
    g1[6] = (int)((((unsigned)(H_ * W_)) & 0xFFFFu) << 16); // stride0 hi | stride1 lo
    g1[7] = (int)(((unsigned)(H_ * W_)) >> 16);             // tensor_dim1_stride[47:16]

    // D# group 2: tensor_dim2=C for z OOB; group 3 unused.
    tdm_i32x4 g2; g2[0] = C_; g2[1] = 0; g2[2] = H_ * W_; g2[3] = 0;
    tdm_i32x4 g3 = {0, 0, 0, 0};

#if defined(__clang_major__) && (__clang_major__ >= 23)
    tdm_i32x8 g4 = {0, 0, 0, 0, 0, 0, 0, 0};
    __builtin_amdgcn_tensor_load_to_lds(g0, g1, g2, g3, g4, 0);
#else
    __builtin_amdgcn_tensor_load_to_lds(g0, g1, g2, g3, 0);
#endif
}

// TOK tokens per block, double-buffered: TDM load of token j+1 overlaps the
// mask-multiply + store of token j. s_wait_tensorcnt(1) waits only for the
// older of the two outstanding tensor DMAs.
__global__ __launch_bounds__(THREADS)
void gms_token_extract(const float* __restrict__ img,
                       const int*   __restrict__ mu,
                       const float* __restrict__ gauss,
                       float*       __restrict__ out)
{
    __shared__ __align__(16) float s_patch[2][ELEMS]; // 2 x 12 KB
    __shared__ __align__(16) float s_gauss[K_ * K_];  // 4 KB

    const int t     = threadIdx.x;
    const int base  = blockIdx.x * TOK;
    const bool w0   = (t < 32);

    // Stage the Gaussian mask (coalesced; L2-hot after the first block).
    #pragma unroll
    for (int i = 0; i < (K_ * K_) / THREADS; ++i)
        s_gauss[t + i * THREADS] = gauss[t + i * THREADS];

    // Prologue: kick off token `base` into buffer 0.
    if (w0) {
        int m0 = __builtin_amdgcn_readfirstlane(mu[2 * base + 0]);
        int m1 = __builtin_amdgcn_readfirstlane(mu[2 * base + 1]);
        tdm_issue(img, base >> 8, m0, m1, (unsigned)(uintptr_t)&s_patch[0][0]);
    }

    #pragma unroll
    for (int j = 0; j < TOK; ++j) {
        const int token = base + j;

        if (w0) {
            if (j + 1 < TOK) {
                const int nt = token + 1;
                int m0 = __builtin_amdgcn_readfirstlane(mu[2 * nt + 0]);
                int m1 = __builtin_amdgcn_readfirstlane(mu[2 * nt + 1]);
                tdm_issue(img, nt >> 8, m0, m1,
                          (unsigned)(uintptr_t)&s_patch[(j + 1) & 1][0]);
                __builtin_amdgcn_s_wait_tensorcnt(1); // wait for token j only
            } else {
                __builtin_amdgcn_s_wait_tensorcnt(0); // last token: drain
            }
        }
        __syncthreads(); // token j's tile visible to all waves

        // float4 drain: ds_load_b128 + global_store_b128, 512 B/wave-inst.
        const float4* p4   = (const float4*)&s_patch[j & 1][0];
        const float4* g4   = (const float4*)&s_gauss[0];
        float4*       dst4 = (float4*)(out + (size_t)token * ELEMS);
        #pragma unroll
        for (int i = 0; i < ELEMS / 4 / THREADS; ++i) { // 3 iterations
            const int f = t + i * THREADS;              // float4 index in tile
            float4 a = p4[f];
            float4 g = g4[f & ((K_ * K_ / 4) - 1)];     // K divisible by 4
            float4 r;
            r.x = a.x * g.x; r.y = a.y * g.y; r.z = a.z * g.z; r.w = a.w * g.w;
            dst4[f] = r;
        }
        __syncthreads(); // all reads of this buffer done before it is re-targeted
    }
}

extern "C" void kernel_launch(void* const* d_in, const int* in_sizes, int n_in,
                              void* d_out, int out_size, void* d_ws, size_t ws_size,
                              hipStream_t stream) {
    (void)in_sizes; (void)n_in; (void)d_ws; (void)ws_size; (void)out_size;
    const float* img   = (const float*)d_in[0];   // (B, C, H, W) fp32
    const int*   mu    = (const int*)d_in[1];     // (B, N, 2) int32
    const float* gauss = (const float*)d_in[2];   // (K, K) fp32
    float*       out   = (float*)d_out;           // (B, N, C*K*K) fp32

    gms_token_extract<<<dim3(NBLK), dim3(THREADS), 0, stream>>>(img, mu, gauss, out);
}